// SpaceTimeElementEncoder_8005819040535
// MI455X (gfx1250) — compile-verified
//
#include <hip/hip_runtime.h>
#include <hip/hip_bf16.h>

#define N_NODES 200000
#define N_EDGES 3200000
#define N_GRAPHS 64
#define IN_DIM 128
#define HIDDEN 16
#define OUT_DIM 64

typedef __attribute__((ext_vector_type(2))) float v2f;
typedef __attribute__((ext_vector_type(8))) float v8f;

// ---------------------------------------------------------------------------
// Degree accumulation: deg[col[e]] += 1 for all edges, plus +1 self loop/node
// ---------------------------------------------------------------------------
__global__ void degree_kernel(const int* __restrict__ col, float* __restrict__ deg,
                              int nE, int nN) {
    int i = blockIdx.x * blockDim.x + threadIdx.x;
    if (i >= nE + nN) return;
    int c = (i < nE) ? col[i] : (i - nE);
    atomicAdd(&deg[c], 1.0f);
}

__global__ void rsqrt_kernel(float* __restrict__ deg, int n) {
    int i = blockIdx.x * blockDim.x + threadIdx.x;
    if (i >= n) return;
    deg[i] = 1.0f / sqrtf(deg[i]);   // deg >= 1 always (self loop)
}

// ---------------------------------------------------------------------------
// GEMM1: h1[200000x16] = x[200000x128] @ W1[128x16], f32 WMMA 16x16x4.
// One wave computes a 16-node x 16-hidden tile; K-loop of 32 WMMA steps.
// A layout (16x4 f32): lanes 0-15 M=0..15, v0=K0 v1=K1; lanes 16-31 v0=K2 v1=K3.
// B layout (4x16 f32): lanes 0-15 N=0..15, v0=K0 v1=K1; lanes 16-31 v0=K2 v1=K3.
// D layout (16x16 f32): vgpr r = row r (lanes 0-15) / row r+8 (lanes 16-31).
// ---------------------------------------------------------------------------
__global__ void gemm1_wmma(const float* __restrict__ x, const float* __restrict__ W1,
                           float* __restrict__ h1) {
    const int lane = threadIdx.x & 31;
    const int wave = threadIdx.x >> 5;
    const int tile = blockIdx.x * 8 + wave;       // 12500 tiles of 16 nodes
    if (tile >= (N_NODES / 16)) return;           // wave-uniform: EXEC stays all-1s
    const int l    = lane & 15;
    const int half = lane >> 4;
    const int nb   = tile * 16;

    v8f c = {0.f, 0.f, 0.f, 0.f, 0.f, 0.f, 0.f, 0.f};
    const float* xr = x + (size_t)(nb + l) * IN_DIM;
#pragma unroll 4
    for (int k = 0; k < IN_DIM; k += 4) {
        v2f a = *(const v2f*)(xr + k + 2 * half);            // x[row][k+2h..k+2h+1]
        v2f b;
        const float* bp = W1 + (size_t)(k + 2 * half) * HIDDEN + l;
        b.x = bp[0];                                          // W1[k+2h][l]
        b.y = bp[HIDDEN];                                     // W1[k+2h+1][l]
        c = __builtin_amdgcn_wmma_f32_16x16x4_f32(false, a, false, b,
                                                  (short)0, c, false, false);
    }
    float* out = h1 + (size_t)(nb + 8 * half) * HIDDEN + l;
#pragma unroll
    for (int r = 0; r < 8; ++r) out[(size_t)r * HIDDEN] = c[r];
}

// ---------------------------------------------------------------------------
// GEMM2: h2[200000x64] = g[200000x16] @ W2[16x64]. 12500 row-tiles x 4 col-tiles.
// ---------------------------------------------------------------------------
__global__ void gemm2_wmma(const float* __restrict__ g, const float* __restrict__ W2,
                           float* __restrict__ h2) {
    const int lane = threadIdx.x & 31;
    const int wave = threadIdx.x >> 5;
    const int t = blockIdx.x * 8 + wave;          // 50000 tiles
    if (t >= (N_NODES / 16) * (OUT_DIM / 16)) return;
    const int rowTile = t >> 2;
    const int colTile = t & 3;
    const int l    = lane & 15;
    const int half = lane >> 4;
    const int nb   = rowTile * 16;
    const int cb   = colTile * 16;

    v8f c = {0.f, 0.f, 0.f, 0.f, 0.f, 0.f, 0.f, 0.f};
    const float* gr = g + (size_t)(nb + l) * HIDDEN;
#pragma unroll
    for (int k = 0; k < HIDDEN; k += 4) {
        v2f a = *(const v2f*)(gr + k + 2 * half);
        v2f b;
        const float* bp = W2 + (size_t)(k + 2 * half) * OUT_DIM + cb + l;
        b.x = bp[0];
        b.y = bp[OUT_DIM];
        c = __builtin_amdgcn_wmma_f32_16x16x4_f32(false, a, false, b,
                                                  (short)0, c, false, false);
    }
    float* out = h2 + (size_t)(nb + 8 * half) * OUT_DIM + cb + l;
#pragma unroll
    for (int r = 0; r < 8; ++r) out[(size_t)r * OUT_DIM] = c[r];
}

// ---------------------------------------------------------------------------
// Edge scatter: out[col] += h[row] * dinv[row]*dinv[col]  (edges + self loops)
// One thread per edge; float4 gathers, f32 global atomics (L2-resident data).
// ---------------------------------------------------------------------------
template <int F>
__global__ void scatter_norm(const float* __restrict__ h, float* __restrict__ out,
                             const int* __restrict__ row, const int* __restrict__ col,
                             const float* __restrict__ dinv, int nE, int nN) {
    int i = blockIdx.x * blockDim.x + threadIdx.x;
    if (i >= nE + nN) return;
    int r, c;
    if (i < nE) { r = row[i]; c = col[i]; }
    else        { r = i - nE; c = r; }
    float w = dinv[r] * dinv[c];
    const float4* hs = (const float4*)(h + (size_t)r * F);
    float* o = out + (size_t)c * F;
#pragma unroll
    for (int q = 0; q < F / 4; ++q) {
        float4 v = hs[q];
        atomicAdd(o + 4 * q + 0, v.x * w);
        atomicAdd(o + 4 * q + 1, v.y * w);
        atomicAdd(o + 4 * q + 2, v.z * w);
        atomicAdd(o + 4 * q + 3, v.w * w);
    }
}

__global__ void bias_relu_kernel(float* __restrict__ g, const float* __restrict__ b1, int n) {
    int i = blockIdx.x * blockDim.x + threadIdx.x;
    if (i >= n) return;
    float v = g[i] + b1[i & (HIDDEN - 1)];
    g[i] = v > 0.0f ? v : 0.0f;
}

// Pool: d_out[batch[node]*64 + f] += out2[node*64+f] + b2[f]
__global__ void pool_kernel(const float* __restrict__ out2, const float* __restrict__ b2,
                            const int* __restrict__ batch, float* __restrict__ dout, int n) {
    int i = blockIdx.x * blockDim.x + threadIdx.x;
    if (i >= n) return;
    int f = i & (OUT_DIM - 1);
    int node = i >> 6;
    atomicAdd(&dout[(size_t)batch[node] * OUT_DIM + f], out2[i] + b2[f]);
}

// ---------------------------------------------------------------------------
extern "C" void kernel_launch(void* const* d_in, const int* in_sizes, int n_in,
                              void* d_out, int out_size, void* d_ws, size_t ws_size,
                              hipStream_t stream) {
    const float* x     = (const float*)d_in[0];
    const int*   ei    = (const int*)d_in[1];
    const int*   batch = (const int*)d_in[2];
    const float* W1    = (const float*)d_in[3];
    const float* b1    = (const float*)d_in[4];
    const float* W2    = (const float*)d_in[5];
    const float* b2    = (const float*)d_in[6];
    float*       out   = (float*)d_out;

    const int* row = ei;
    const int* col = ei + N_EDGES;

    // Workspace layout (bytes):
    //   [0, 51.2M)      h1 (12.8 MB) then h2 (51.2 MB) -- h1 dead before h2 write
    //   [51.2M, 64M)    out1 / g (12.8 MB)
    //   [64M, 64.8M)    deg -> dinv (0.8 MB)
    //   [64.8M, 116M)   out2 (51.2 MB)
    char*  ws   = (char*)d_ws;
    float* h12  = (float*)(ws);
    float* g    = (float*)(ws + 51200000);
    float* dinv = (float*)(ws + 64000000);
    float* out2 = (float*)(ws + 64800000);

    hipMemsetAsync(dinv, 0, (size_t)N_NODES * 4, stream);
    hipMemsetAsync(g,    0, (size_t)N_NODES * HIDDEN * 4, stream);
    hipMemsetAsync(out2, 0, (size_t)N_NODES * OUT_DIM * 4, stream);
    hipMemsetAsync(out,  0, (size_t)N_GRAPHS * OUT_DIM * 4, stream);

    const int totalEN = N_EDGES + N_NODES;
    const int B = 256;

    degree_kernel<<<(totalEN + B - 1) / B, B, 0, stream>>>(col, dinv, N_EDGES, N_NODES);
    rsqrt_kernel<<<(N_NODES + B - 1) / B, B, 0, stream>>>(dinv, N_NODES);

    // GEMM1: 12500 tiles, 8 waves/block
    gemm1_wmma<<<(12500 + 7) / 8, B, 0, stream>>>(x, W1, h12);

    scatter_norm<HIDDEN><<<(totalEN + B - 1) / B, B, 0, stream>>>(h12, g, row, col,
                                                                  dinv, N_EDGES, N_NODES);
    bias_relu_kernel<<<(N_NODES * HIDDEN + B - 1) / B, B, 0, stream>>>(g, b1,
                                                                       N_NODES * HIDDEN);

    // GEMM2: 50000 tiles, 8 waves/block
    gemm2_wmma<<<50000 / 8, B, 0, stream>>>(g, W2, h12);

    scatter_norm<OUT_DIM><<<(totalEN + B - 1) / B, B, 0, stream>>>(h12, out2, row, col,
                                                                   dinv, N_EDGES, N_NODES);
    pool_kernel<<<(N_NODES * OUT_DIM + B - 1) / B, B, 0, stream>>>(out2, b2, batch, out,
                                                                   N_NODES * OUT_DIM);
}